// NeuroGameTransformer_75127567942223
// MI455X (gfx1250) — compile-verified
//
#include <hip/hip_runtime.h>
#include <hip/hip_bf16.h>
#include <math.h>

// ---------------------------------------------------------------------------
// NeuroGameTransformer for gfx1250 (MI455X).
// Kernel A: fused [cs = coalitions@feats] -> [proj = cs@W_v] row-norms via
//           V_WMMA_F32_16X16X4_F32 with W_v K-slabs staged through LDS
//           (software-pipelined via registers) -> shapley. One block/batch.
// Kernel B: pooled projection -> 25-step mean-field scan -> free energy ->
//           LayerNorm/GELU MLP head -> logits. One block per batch.
// ---------------------------------------------------------------------------

typedef float v2f __attribute__((ext_vector_type(2)));
typedef float v4f __attribute__((ext_vector_type(4)));
typedef float v8f __attribute__((ext_vector_type(8)));

#define DD   768
#define NN   128
#define KK   25
#define BB   256

// LDS layout for kernel A (floats)
#define CS_STRIDE 772                     // 768+4: A rows land on distinct banks
#define B_STRIDE  776                     // 2*776 % 64 = 16: lane halves disjoint
#define CS_FLOATS (32 * CS_STRIDE)        // 24704
#define COAL_OFF  CS_FLOATS               // 24704
#define B_OFF     (COAL_OFF + KK*NN)      // 27904
#define B_FLOATS  (16 * B_STRIDE)         // 12416
#define ROWSQ_OFF (B_OFF + B_FLOATS)      // 40320
#define V_OFF     (ROWSQ_OFF + 32)        // 40352
#define SMEMA_FLOATS (V_OFF + 32)         // 40384 floats = 161536 bytes

__global__ __launch_bounds__(256) void shapley_wmma_kernel(
    const float* __restrict__ hs,      // (256,128,768)
    const float* __restrict__ coal,    // (25,128)
    const float* __restrict__ Wv,      // (768,768)
    const float* __restrict__ scale,   // (1,)
    float* __restrict__ out)           // d_out base (shapley at offset 1024)
{
    extern __shared__ float smem[];
    float* s_cs    = smem;              // [32][772]
    float* s_coal  = smem + COAL_OFF;   // [25*128]
    float* s_b     = smem + B_OFF;      // [16][776] staged W_v K-slab
    float* s_rowsq = smem + ROWSQ_OFF;  // [32]
    float* s_v     = smem + V_OFF;      // [32]

    const int b   = blockIdx.x;
    const int tid = threadIdx.x;

    // --- stage coalitions, zero reduction slots, zero pad rows 25..31 -------
    for (int i = tid; i < KK * NN; i += 256) s_coal[i] = coal[i];
    if (tid < 32) s_rowsq[tid] = 0.0f;
    for (int i = tid; i < 7 * DD; i += 256) {
        int r = 25 + i / DD;
        int c = i % DD;
        s_cs[r * CS_STRIDE + c] = 0.0f;
    }
    __syncthreads();

    // --- phase 1: cs[k, d] = sum_n coal[k,n] * hs[b,n,d] (25 x 768 in LDS) --
    #pragma unroll 1
    for (int c = 0; c < 3; ++c) {
        const int d = c * 256 + tid;
        float acc[KK];
        #pragma unroll
        for (int k = 0; k < KK; ++k) acc[k] = 0.0f;
        const float* hrow = hs + (size_t)b * NN * DD + d;
        #pragma unroll 1
        for (int n = 0; n < NN; ++n) {
            const float x = hrow[n * DD];
            #pragma unroll
            for (int k = 0; k < KK; ++k)
                acc[k] = fmaf(s_coal[k * NN + n], x, acc[k]);
        }
        #pragma unroll
        for (int k = 0; k < KK; ++k) s_cs[k * CS_STRIDE + d] = acc[k];
    }

    // --- phase 2: row sum-of-squares of proj = cs @ Wv via f32 WMMA ---------
    // W_v is consumed in K=16 slabs staged into LDS once per block; the next
    // slab is prefetched into registers while the current slab feeds WMMA.
    const int wave  = tid >> 5;
    const int lane  = tid & 31;
    const int mtile = wave & 1;           // rows 16*mtile .. +15
    const int nt0   = wave >> 1;          // ntile = nt0 + 4*ti, ti = 0..11
    const int lh    = lane >> 4;          // lane half
    const int ll    = lane & 15;
    const int row   = mtile * 16 + ll;    // A-matrix row for this lane
    const int koff  = 2 * lh;             // K offset pattern of 16x16x4 layout

    v8f acc[12];
    #pragma unroll
    for (int ti = 0; ti < 12; ++ti)
        acc[ti] = (v8f){0.f, 0.f, 0.f, 0.f, 0.f, 0.f, 0.f, 0.f};

    // per-thread staging slice: 12 float4 = 48 floats of the 16x768 slab
    v4f breg[12];
    {
        #pragma unroll
        for (int i = 0; i < 12; ++i) {
            const int idx = tid + i * 256;        // 0..3071
            const int r   = idx / 192;            // slab row 0..15
            const int c4  = idx % 192;            // float4 column
            breg[i] = *(const v4f*)(Wv + (size_t)r * DD + c4 * 4);
        }
    }

    #pragma unroll 1
    for (int kc = 0; kc < DD; kc += 16) {
        __syncthreads();                          // slab consumers done
        #pragma unroll
        for (int i = 0; i < 12; ++i) {
            const int idx = tid + i * 256;
            const int r   = idx / 192;
            const int c4  = idx % 192;
            *(v4f*)(s_b + r * B_STRIDE + c4 * 4) = breg[i];
        }
        __syncthreads();                          // slab visible

        if (kc + 16 < DD) {                       // prefetch next slab
            #pragma unroll
            for (int i = 0; i < 12; ++i) {
                const int idx = tid + i * 256;
                const int r   = idx / 192;
                const int c4  = idx % 192;
                breg[i] = *(const v4f*)(Wv + (size_t)(kc + 16 + r) * DD + c4 * 4);
            }
        }

        const float* arow = s_cs + row * CS_STRIDE + kc + koff;
        #pragma unroll
        for (int ti = 0; ti < 12; ++ti) {
            const int ntile = nt0 + ti * 4;
            const float* bcol = s_b + koff * B_STRIDE + ntile * 16 + ll;
            #pragma unroll
            for (int kb = 0; kb < 16; kb += 4) {
                // A: cs[row, kc+kb+koff .. +1]   (8B-aligned LDS pair)
                v2f a = *(const v2f*)(arow + kb);
                // B: Wv[kc+kb+koff, col], Wv[kc+kb+koff+1, col] from LDS slab
                v2f bb;
                bb.x = bcol[kb * B_STRIDE];
                bb.y = bcol[(kb + 1) * B_STRIDE];
                acc[ti] = __builtin_amdgcn_wmma_f32_16x16x4_f32(
                    false, a, false, bb, (short)0, acc[ti], false, false);
            }
        }
    }

    // per-lane sum of squares over this wave's 12 output tiles
    float locsq[8];
    #pragma unroll
    for (int v = 0; v < 8; ++v) locsq[v] = 0.0f;
    #pragma unroll
    for (int ti = 0; ti < 12; ++ti) {
        #pragma unroll
        for (int v = 0; v < 8; ++v)
            locsq[v] = fmaf(acc[ti][v], acc[ti][v], locsq[v]);
    }
    // D layout: vgpr v -> row mtile*16 + v + 8*lanehalf
    {
        const int rbase = mtile * 16 + 8 * lh;
        #pragma unroll
        for (int v = 0; v < 8; ++v)
            atomicAdd(&s_rowsq[rbase + v], locsq[v]);   // ds_add_f32
    }
    __syncthreads();

    // --- phase 3: v = relu(||proj_k|| * scale), then shapley ----------------
    if (tid < KK)
        s_v[tid] = fmaxf(sqrtf(s_rowsq[tid]) * scale[0], 0.0f);
    __syncthreads();

    if (tid < NN) {
        float cnt1 = 0.0f, sw = 0.0f, sv = 0.0f;
        #pragma unroll
        for (int k = 0; k < KK; ++k) {
            const float c  = s_coal[k * NN + tid];
            const float vk = s_v[k];
            cnt1 += c;
            sw  = fmaf(vk, c, sw);
            sv += vk;
        }
        const float cnt0  = (float)KK - cnt1;
        const bool  valid = (cnt1 > 0.0f) && (cnt0 > 0.0f);
        const float res = valid
            ? sw / fmaxf(cnt1, 1.0f) - (sv - sw) / fmaxf(cnt0, 1.0f)
            : 0.0f;
        out[1024 + b * NN + tid] = res;   // shapley block of d_out
    }
}

// ---------------------------------------------------------------------------

__device__ __forceinline__ float gelu_exact(float x) {
    return 0.5f * x * (1.0f + erff(x * 0.70710678118654752f));
}

__global__ __launch_bounds__(128) void meanfield_head_kernel(
    const float* __restrict__ hs,   // (256,128,768) -> pooled = hs[b,0,:]
    const float* __restrict__ lf,   // (128,)
    const float* __restrict__ J,    // (128,128)
    const float* __restrict__ Wp,   // (768,128)
    const float* __restrict__ bp,   // (128,)
    const float* __restrict__ W1,   // (128,256)
    const float* __restrict__ b1, const float* __restrict__ g1, const float* __restrict__ be1,
    const float* __restrict__ W2,   // (256,128)
    const float* __restrict__ b2, const float* __restrict__ g2, const float* __restrict__ be2,
    const float* __restrict__ W3,   // (128,3)
    const float* __restrict__ b3,
    float* __restrict__ out)        // logits at 0, free_energy at 768
{
    const int b = blockIdx.x;
    const int t = threadIdx.x;

    __shared__ float sp[DD];
    __shared__ float sm[NN];
    __shared__ float red[NN];
    __shared__ float sx1[256];
    __shared__ float sx2[NN];

    for (int i = t; i < DD; i += NN) sp[i] = hs[(size_t)b * NN * DD + i];
    __syncthreads();

    // m0 = tanh((pooled @ Wp + bp) * 0.1)
    float acc = bp[t];
    #pragma unroll 4
    for (int d = 0; d < DD; ++d) acc = fmaf(sp[d], Wp[d * NN + t], acc);
    sm[t] = tanhf(acc * 0.1f);
    __syncthreads();

    // 25 damped mean-field iterations; Jm = J with zero diagonal
    #pragma unroll 1
    for (int it = 0; it < 25; ++it) {
        float h = lf[t];
        #pragma unroll 4
        for (int n = 0; n < NN; ++n) {
            const float jv = (n == t) ? 0.0f : J[n * NN + t];
            h = fmaf(sm[n], jv, h);
        }
        __syncthreads();
        sm[t] = fmaf(0.7f, tanhf(h * 4.0f), 0.3f * sm[t]);
        __syncthreads();
    }

    // free energy: -gamma * sum logaddexp(h/g, -h/g), gamma = 0.25
    {
        float h = lf[t];
        #pragma unroll 4
        for (int n = 0; n < NN; ++n) {
            const float jv = (n == t) ? 0.0f : J[n * NN + t];
            h = fmaf(sm[n], jv, h);
        }
        const float a = fabsf(h * 4.0f);
        red[t] = a + log1pf(expf(-2.0f * a));
    }
    __syncthreads();
    for (int s = 64; s > 0; s >>= 1) { if (t < s) red[t] += red[t + s]; __syncthreads(); }
    if (t == 0) out[768 + b] = -0.25f * red[0];
    __syncthreads();

    // layer 1: x1 = gelu(LN(m @ W1 + b1)); 256 outputs -> 2 per thread
    float xa = b1[t], xb = b1[t + NN];
    #pragma unroll 4
    for (int n = 0; n < NN; ++n) {
        const float mv = sm[n];
        xa = fmaf(mv, W1[n * 256 + t], xa);
        xb = fmaf(mv, W1[n * 256 + t + NN], xb);
    }
    red[t] = xa + xb;
    __syncthreads();
    for (int s = 64; s > 0; s >>= 1) { if (t < s) red[t] += red[t + s]; __syncthreads(); }
    const float mean1 = red[0] * (1.0f / 256.0f);
    __syncthreads();
    const float da = xa - mean1, db = xb - mean1;
    red[t] = da * da + db * db;
    __syncthreads();
    for (int s = 64; s > 0; s >>= 1) { if (t < s) red[t] += red[t + s]; __syncthreads(); }
    const float inv1 = rsqrtf(red[0] * (1.0f / 256.0f) + 1e-5f);
    __syncthreads();
    sx1[t]      = gelu_exact(fmaf(da * inv1, g1[t],      be1[t]));
    sx1[t + NN] = gelu_exact(fmaf(db * inv1, g1[t + NN], be1[t + NN]));
    __syncthreads();

    // layer 2: x2 = gelu(LN(x1 @ W2 + b2)); 128 outputs
    float x2 = b2[t];
    #pragma unroll 4
    for (int j = 0; j < 256; ++j) x2 = fmaf(sx1[j], W2[j * NN + t], x2);
    red[t] = x2;
    __syncthreads();
    for (int s = 64; s > 0; s >>= 1) { if (t < s) red[t] += red[t + s]; __syncthreads(); }
    const float mean2 = red[0] * (1.0f / 128.0f);
    __syncthreads();
    const float d2 = x2 - mean2;
    red[t] = d2 * d2;
    __syncthreads();
    for (int s = 64; s > 0; s >>= 1) { if (t < s) red[t] += red[t + s]; __syncthreads(); }
    const float inv2 = rsqrtf(red[0] * (1.0f / 128.0f) + 1e-5f);
    __syncthreads();
    sx2[t] = gelu_exact(fmaf(d2 * inv2, g2[t], be2[t]));
    __syncthreads();

    // logits = x2 @ W3 + b3 (3 outputs)
    if (t < 3) {
        float l = b3[t];
        #pragma unroll 4
        for (int j = 0; j < NN; ++j) l = fmaf(sx2[j], W3[j * 3 + t], l);
        out[b * 3 + t] = l;
    }
}

// ---------------------------------------------------------------------------

extern "C" void kernel_launch(void* const* d_in, const int* in_sizes, int n_in,
                              void* d_out, int out_size, void* d_ws, size_t ws_size,
                              hipStream_t stream) {
    (void)in_sizes; (void)n_in; (void)out_size; (void)d_ws; (void)ws_size;

    const float* hs    = (const float*)d_in[0];
    const float* coalp = (const float*)d_in[1];
    const float* Wv    = (const float*)d_in[2];
    const float* scale = (const float*)d_in[3];
    const float* lf    = (const float*)d_in[4];
    const float* J     = (const float*)d_in[5];
    const float* Wp    = (const float*)d_in[6];
    const float* bp    = (const float*)d_in[7];
    const float* W1    = (const float*)d_in[8];
    const float* b1    = (const float*)d_in[9];
    const float* g1    = (const float*)d_in[10];
    const float* be1   = (const float*)d_in[11];
    const float* W2    = (const float*)d_in[12];
    const float* b2    = (const float*)d_in[13];
    const float* g2    = (const float*)d_in[14];
    const float* be2   = (const float*)d_in[15];
    const float* W3    = (const float*)d_in[16];
    const float* b3    = (const float*)d_in[17];
    float* out = (float*)d_out;

    const size_t smemA = SMEMA_FLOATS * sizeof(float);   // ~158 KB dynamic LDS
    shapley_wmma_kernel<<<BB, 256, smemA, stream>>>(hs, coalp, Wv, scale, out);
    meanfield_head_kernel<<<BB, 128, 0, stream>>>(hs, lf, J, Wp, bp,
                                                  W1, b1, g1, be1,
                                                  W2, b2, g2, be2,
                                                  W3, b3, out);
}